// DiffeomorphicTransform_12996571038076
// MI455X (gfx1250) — compile-verified
//
#include <hip/hip_runtime.h>

// Diffeomorphic transform via scaling-and-squaring, MI455X (gfx1250).
// flow: [2,3,160,160,160] f32. 7 iterations of flow += trilinear(flow, grid+flow).
// Memory-bound gather workload: no dense matmul -> WMMA/TDM not applicable.
// Strategy: interleaved [...,3] internal layout (per-corner channels contiguous
// -> merged gather loads), ping-pong between d_ws and d_out (each buffer 98MB,
// fits the 192MB L2), wave32-friendly 256-thread blocks, one voxel per lane.

#define WD   160
#define NVOX (WD * WD * WD)       // 4,096,000
#define NB   2
#define NTOT (NB * NVOX)          // 8,192,000
#define SCL  79.5f                // 0.5f * (WD - 1): align_corners unnormalize

// ---------------------------------------------------------------------------
// init: planar input -> interleaved, scaled by 2^-7
// ---------------------------------------------------------------------------
__global__ __launch_bounds__(256) void dt_init(const float* __restrict__ in,
                                               float* __restrict__ dst) {
  int idx = blockIdx.x * blockDim.x + threadIdx.x;
  if (idx >= NTOT) return;
  int b = idx / NVOX;
  int v = idx - b * NVOX;
  const float s = 1.0f / 128.0f;          // 2^-TIME_STEP
  int pb = b * 3 * NVOX;
  float f0 = in[pb + 0 * NVOX + v] * s;   // channel 0 -> z (D)
  float f1 = in[pb + 1 * NVOX + v] * s;   // channel 1 -> y (H)
  float f2 = in[pb + 2 * NVOX + v] * s;   // channel 2 -> x (W)
  int o = idx * 3;
  dst[o + 0] = f0;
  dst[o + 1] = f1;
  dst[o + 2] = f2;
}

// ---------------------------------------------------------------------------
// one scaling-and-squaring step: dst = src + trilinear_border(src, grid + src)
// src interleaved; dst interleaved, or planar f32 (final step into d_out)
// ---------------------------------------------------------------------------
template <bool PLANAR_OUT>
__global__ __launch_bounds__(256) void dt_step(const float* __restrict__ src,
                                               float* __restrict__ dst) {
  int idx = blockIdx.x * blockDim.x + threadIdx.x;
  if (idx >= NTOT) return;
  int b = idx / NVOX;
  int v = idx - b * NVOX;
  int x = v % WD;
  int t = v / WD;
  int y = t % WD;
  int z = t / WD;

  int so = idx * 3;
  float f0 = src[so + 0];
  float f1 = src[so + 1];
  float f2 = src[so + 2];

  // sample position in voxel units, clamped to border (align_corners=True)
  float px = fminf(fmaxf((float)x + f2 * SCL, 0.0f), (float)(WD - 1));
  float py = fminf(fmaxf((float)y + f1 * SCL, 0.0f), (float)(WD - 1));
  float pz = fminf(fmaxf((float)z + f0 * SCL, 0.0f), (float)(WD - 1));

  int x0 = (int)px, y0 = (int)py, z0 = (int)pz;   // px>=0 -> trunc == floor
  int x1 = min(x0 + 1, WD - 1);
  int y1 = min(y0 + 1, WD - 1);
  int z1 = min(z0 + 1, WD - 1);
  float wx = px - (float)x0, wy = py - (float)y0, wz = pz - (float)z0;
  float ox = 1.0f - wx, oy = 1.0f - wy, oz = 1.0f - wz;

  int bb  = b * NVOX;
  int r00 = (z0 * WD + y0) * WD;
  int r01 = (z0 * WD + y1) * WD;
  int r10 = (z1 * WD + y0) * WD;
  int r11 = (z1 * WD + y1) * WD;

  const float* p000 = src + (bb + r00 + x0) * 3;
  const float* p001 = src + (bb + r00 + x1) * 3;
  const float* p010 = src + (bb + r01 + x0) * 3;
  const float* p011 = src + (bb + r01 + x1) * 3;
  const float* p100 = src + (bb + r10 + x0) * 3;
  const float* p101 = src + (bb + r10 + x1) * 3;
  const float* p110 = src + (bb + r11 + x0) * 3;
  const float* p111 = src + (bb + r11 + x1) * 3;

  float w000 = oz * oy * ox, w001 = oz * oy * wx;
  float w010 = oz * wy * ox, w011 = oz * wy * wx;
  float w100 = wz * oy * ox, w101 = wz * oy * wx;
  float w110 = wz * wy * ox, w111 = wz * wy * wx;

  float s0 = p000[0] * w000 + p001[0] * w001 + p010[0] * w010 + p011[0] * w011
           + p100[0] * w100 + p101[0] * w101 + p110[0] * w110 + p111[0] * w111;
  float s1 = p000[1] * w000 + p001[1] * w001 + p010[1] * w010 + p011[1] * w011
           + p100[1] * w100 + p101[1] * w101 + p110[1] * w110 + p111[1] * w111;
  float s2 = p000[2] * w000 + p001[2] * w001 + p010[2] * w010 + p011[2] * w011
           + p100[2] * w100 + p101[2] * w101 + p110[2] * w110 + p111[2] * w111;

  float r0 = f0 + s0;
  float r1 = f1 + s1;
  float r2 = f2 + s2;

  if (PLANAR_OUT) {
    int pbo = b * 3 * NVOX;
    dst[pbo + 0 * NVOX + v] = r0;
    dst[pbo + 1 * NVOX + v] = r1;
    dst[pbo + 2 * NVOX + v] = r2;
  } else {
    dst[so + 0] = r0;
    dst[so + 1] = r1;
    dst[so + 2] = r2;
  }
}

// ---------------------------------------------------------------------------
// launch: init -> 6 interleaved ping-pong steps -> final planar step
//   A = d_ws   (interleaved scratch, 2*160^3*3*4 = 98.3 MB required)
//   B = d_out  (reused as interleaved scratch; final step overwrites planar)
// ---------------------------------------------------------------------------
extern "C" void kernel_launch(void* const* d_in, const int* in_sizes, int n_in,
                              void* d_out, int out_size, void* d_ws, size_t ws_size,
                              hipStream_t stream) {
  (void)in_sizes; (void)n_in; (void)out_size; (void)ws_size;
  const float* in = (const float*)d_in[0];
  float* A = (float*)d_ws;
  float* B = (float*)d_out;

  const int threads = 256;
  const int blocks = (NTOT + threads - 1) / threads;

  dt_init<<<blocks, threads, 0, stream>>>(in, A);          // A = in / 2^7
  dt_step<false><<<blocks, threads, 0, stream>>>(A, B);    // step 1
  dt_step<false><<<blocks, threads, 0, stream>>>(B, A);    // step 2
  dt_step<false><<<blocks, threads, 0, stream>>>(A, B);    // step 3
  dt_step<false><<<blocks, threads, 0, stream>>>(B, A);    // step 4
  dt_step<false><<<blocks, threads, 0, stream>>>(A, B);    // step 5
  dt_step<false><<<blocks, threads, 0, stream>>>(B, A);    // step 6
  dt_step<true ><<<blocks, threads, 0, stream>>>(A, (float*)d_out); // step 7 -> planar
}